// ConfidanceLoss_32401233281640
// MI455X (gfx1250) — compile-verified
//
#include <hip/hip_runtime.h>
#include <hip/hip_bf16.h>

// Problem constants from the reference
#define GRID   32
#define DNUM   8                       // cells per axis
#define PCELLS 512                     // DNUM^3
#define VOLN   (GRID * GRID * GRID)    // 32768 voxels per batch
#define ROWPAD 36                      // padded LDS row stride in ints (144B, 16B-aligned, breaks bank conflicts)
#define CHROWS 256                     // rows staged per chunk (4 chunks x 256 = 1024 rows)

typedef __attribute__((ext_vector_type(4))) int v4i;
typedef __attribute__((address_space(1))) v4i  g_v4i;   // global
typedef __attribute__((address_space(3))) v4i  l_v4i;   // LDS

#if __has_builtin(__builtin_amdgcn_global_load_async_to_lds_b128)
#define HAVE_ASYNC_LDS 1
#else
#define HAVE_ASYNC_LDS 0
#endif

__device__ __forceinline__ void wait_asynccnt0() {
#if __has_builtin(__builtin_amdgcn_s_wait_asynccnt)
    __builtin_amdgcn_s_wait_asynccnt(0);
#else
    asm volatile("s_wait_asynccnt 0" ::: "memory");
#endif
}

extern "C" __global__ void __launch_bounds__(256)
confidance_loss_kernel(const float* __restrict__ confi,
                       const int*   __restrict__ vol,
                       float*       __restrict__ out,
                       int B)
{
    const int b = blockIdx.x;
    const int t = threadIdx.x;
    const size_t BP = (size_t)B * PCELLS;

    __shared__ __align__(16) int raw[CHROWS * ROWPAD];   // 36864 B: one chunk of padded rows
    __shared__ unsigned char m3[GRID * GRID * DNUM];     //  8192 B: dim3-reduced maxima
    __shared__ unsigned char m23[GRID * DNUM * DNUM];    //  2048 B: dim2-reduced maxima

    const int* volB = vol + (size_t)b * VOLN;

    // ---- Pass 1 (chunked): async-stage rows into LDS (CDNA5 ASYNCcnt path),
    //      then reduce contiguous dim3 into 8 window maxima per row.
    for (int c = 0; c < 4; ++c) {
        if (c) __syncthreads();   // previous chunk fully consumed before overwrite

        // Stage 256 rows = 2048 int4; 8 int4 per thread, coalesced per wave.
#pragma unroll
        for (int k = 0; k < 8; ++k) {
            const int g4 = k * 256 + t;      // int4 index within chunk
            const int rr = g4 >> 3;          // row within chunk
            const int k4 = g4 & 7;           // int4 within row
#if HAVE_ASYNC_LDS
            __builtin_amdgcn_global_load_async_to_lds_b128(
                (g_v4i*)(volB + (size_t)(c * 2048 + g4) * 4),
                (l_v4i*)(raw + rr * ROWPAD + k4 * 4),
                0, 0);
#else
            const int4 x = *(const int4*)(volB + (size_t)(c * 2048 + g4) * 4);
            *(int4*)(raw + rr * ROWPAD + k4 * 4) = x;
#endif
        }
#if HAVE_ASYNC_LDS
        wait_asynccnt0();        // this wave's async copies have landed in LDS
#endif
        __syncthreads();         // all waves' copies visible

        // One padded row per thread (CHROWS == blockDim.x).
        {
            const int r = c * CHROWS + t;    // global row id = i1*32 + i2
            int v[GRID];
#pragma unroll
            for (int q = 0; q < 8; ++q) {
                const int4 x = ((const int4*)(raw + t * ROWPAD))[q];
                v[4 * q + 0] = x.x; v[4 * q + 1] = x.y;
                v[4 * q + 2] = x.z; v[4 * q + 3] = x.w;
            }
#pragma unroll
            for (int c3 = 0; c3 < DNUM; ++c3) {
                const int lo = 4 * c3;
                const int hi = (c3 == DNUM - 1) ? (GRID - 1) : (4 * c3 + 4);
                int m = 0;
#pragma unroll
                for (int i = 0; i < 5; ++i) {
                    const int idx = lo + i;
                    if (idx <= hi) m |= (v[idx] == 1) ? 1 : 0;
                }
                m3[r * DNUM + c3] = (unsigned char)m;
            }
        }
    }
    __syncthreads();

    // ---- Pass 2: reduce dim2 windows -> m23[i1][c2][c3] (2048 values)
#pragma unroll
    for (int g = 0; g < 8; ++g) {
        const int o  = g * 256 + t;
        const int i1 = o >> 6;
        const int c2 = (o >> 3) & 7;
        const int c3 = o & 7;
        const int lo = 4 * c2;
        const int hi = (c2 == DNUM - 1) ? (GRID - 1) : (4 * c2 + 4);
        int m = 0;
        for (int i2 = lo; i2 <= hi; ++i2)
            m |= m3[(i1 * GRID + i2) * DNUM + c3];
        m23[o] = (unsigned char)m;
    }
    __syncthreads();

    // ---- Pass 3: reduce dim1 windows; output flatten j = c3*64 + c2*8 + c1.
    float* out_confi = out;
    float* out_iou   = out + BP;
    float* out_use   = out + 2 * BP;
#pragma unroll
    for (int g = 0; g < 2; ++g) {
        const int j  = g * 256 + t;
        const int c1 = j & 7;
        const int c2 = (j >> 3) & 7;
        const int c3 = j >> 6;
        const int lo = 4 * c1;
        const int hi = (c1 == DNUM - 1) ? (GRID - 1) : (4 * c1 + 4);
        int m = 0;
        for (int i1 = lo; i1 <= hi; ++i1)
            m |= m23[i1 * 64 + c2 * 8 + c3];
        const float f = (float)m;
        const size_t idx = (size_t)b * PCELLS + j;
        out_iou[idx]   = f;         // iou_gt in {0.0, 1.0}
        out_use[idx]   = f;         // in_use == int(iou_gt), emitted as float
        out_confi[idx] = confi[idx];// squeeze(confi_rlt)
    }
}

extern "C" void kernel_launch(void* const* d_in, const int* in_sizes, int n_in,
                              void* d_out, int out_size, void* d_ws, size_t ws_size,
                              hipStream_t stream) {
    // Input order: shape_rlt, trans_rlt, quat_rlt, confi_rlt, batchVolume.
    const float* confi = (const float*)d_in[3];
    const int*   vol   = (const int*)d_in[4];
    float*       out   = (float*)d_out;
    const int B = in_sizes[4] / VOLN;   // 1024 for the reference sizes
    confidance_loss_kernel<<<B, 256, 0, stream>>>(confi, vol, out, B);
    (void)d_ws; (void)ws_size; (void)n_in; (void)out_size;
}